// Attention_24653112279197
// MI455X (gfx1250) — compile-verified
//
#include <hip/hip_runtime.h>
#include <math.h>

// Problem shape (from reference)
#define B_  8
#define EL  512   // ENC_LEN
#define DL  256   // DEC_LEN
#define EH  512   // ENC_H == DEC_H
#define AT  128   // ATTN

typedef __attribute__((ext_vector_type(2))) float v2f;
typedef __attribute__((ext_vector_type(8))) float v8f;

// gfx1250 has a hardware tanh transcendental (v_tanh_f32). Use it if the
// toolchain exposes a builtin; otherwise fall back to libm.
#if __has_builtin(__builtin_amdgcn_tanhf)
#define TANHF(x) __builtin_amdgcn_tanhf(x)
#elif __has_builtin(__builtin_amdgcn_tanh_f32)
#define TANHF(x) __builtin_amdgcn_tanh_f32(x)
#else
#define TANHF(x) tanhf(x)
#endif

// ---------------------------------------------------------------------------
// Stage 1: projection GEMM   D[M, 128] = X[M, 512] @ W[128, 512]^T (+ bias)
// One wave32 computes one 16x16 tile via v_wmma_f32_16x16x4_f32, K-loop 512/4.
// A lane layout (16x4 f32): lane l -> row m=l&15, K pair k+2*(l>>4)+{0,1}
// B lane layout (4x16 f32): lane l -> col n=l&15, same K pair. Since
// B[k][n] = W[n][k], both A and B loads are contiguous float2 from row-major.
// C/D: vgpr i holds (row0 + i + 8*(l>>4), col0 + (l&15)).
// ---------------------------------------------------------------------------
__global__ void proj_gemm_kernel(const float* __restrict__ X,
                                 const float* __restrict__ W,
                                 const float* __restrict__ bias,
                                 float* __restrict__ D,
                                 int n_tiles)
{
    const int wave = threadIdx.x >> 5;
    const int lane = threadIdx.x & 31;
    const int tile = blockIdx.x * (blockDim.x >> 5) + wave;
    if (tile >= n_tiles) return;              // whole-wave uniform exit

    const int mt   = tile >> 3;               // 128/16 = 8 n-tiles
    const int nt   = tile & 7;
    const int row0 = mt << 4;
    const int col0 = nt << 4;
    const int m    = lane & 15;               // row within tile (also col for B)
    const int hi   = lane >> 4;               // K-pair selector

    const float* Xrow = X + (size_t)(row0 + m) * EH;
    const float* Wrow = W + (size_t)(col0 + m) * EH;

    v8f acc = {0.f, 0.f, 0.f, 0.f, 0.f, 0.f, 0.f, 0.f};
#pragma unroll 4
    for (int k0 = 0; k0 < EH; k0 += 4) {
        const int kk = k0 + (hi << 1);
        v2f a = *(const v2f*)(Xrow + kk);
        v2f b = *(const v2f*)(Wrow + kk);
        acc = __builtin_amdgcn_wmma_f32_16x16x4_f32(
            /*neg_a=*/false, a, /*neg_b=*/false, b,
            /*c_mod=*/(short)0, acc, /*reuse_a=*/false, /*reuse_b=*/false);
    }

    const float bv = bias ? bias[col0 + m] : 0.f;
#pragma unroll
    for (int i = 0; i < 8; ++i) {
        const int r = row0 + i + (hi << 3);
        D[(size_t)r * AT + col0 + m] = acc[i] + bv;
    }
}

// ---------------------------------------------------------------------------
// Stage 2: fused additive-attention scores + masked softmax.
// Block = 512 threads (16 wave32s), one thread per encoder position s.
// Whole proj_enc[b] (512x128 f32) staged into LDS (~264KB of the 320KB WGP
// LDS) with gfx1250 async-to-LDS copies (ASYNCcnt), reused across TCHUNK
// decoder positions. Row stride padded to 132 floats to avoid bank conflicts.
// softmax∘mask∘renorm collapses to: w = exp(x-max)*mask / sum(exp(x-max)*mask)
// ---------------------------------------------------------------------------
#define TCHUNK    8
#define PE_STRIDE 132  // 128 + 4 pad; keeps float4 16B alignment
#define NWAVES    (EL / 32)   // 16 waves per block

#define SMEM_FLOATS (EL * PE_STRIDE + TCHUNK * AT + AT + NWAVES)
#define SMEM_BYTES  (SMEM_FLOATS * 4)

__device__ __forceinline__ float wave_max(float x) {
#pragma unroll
    for (int off = 16; off > 0; off >>= 1)
        x = fmaxf(x, __shfl_xor(x, off, 32));
    return x;
}
__device__ __forceinline__ float wave_sum(float x) {
#pragma unroll
    for (int off = 16; off > 0; off >>= 1)
        x += __shfl_xor(x, off, 32);
    return x;
}

__global__ void attn_score_softmax_kernel(const float* __restrict__ pe,   // [B,EL,AT]
                                          const float* __restrict__ pd,   // [B,DL,AT]
                                          const float* __restrict__ vvec, // [AT]
                                          const float* __restrict__ mask, // [B,EL]
                                          float* __restrict__ attn)       // [B,DL,EL]
{
    extern __shared__ float smem[];
    // Dynamic LDS starts at offset 0 (no static LDS in this kernel), so the
    // async-copy VDST offset below is just the float index * 4.
    float* s_pe  = smem;                        // EL * PE_STRIDE
    float* s_pd  = s_pe + EL * PE_STRIDE;       // TCHUNK * AT
    float* s_v   = s_pd + TCHUNK * AT;          // AT
    float* s_red = s_v + AT;                    // NWAVES (cross-wave scratch)

    const int tid  = threadIdx.x;               // == encoder position s
    const int wv   = tid >> 5;
    const int lane = tid & 31;
    const int b    = blockIdx.x >> 5;           // DL/TCHUNK = 32 chunks per batch
    const int t0   = (blockIdx.x & 31) * TCHUNK;

    // Async stage of proj_enc[b] into LDS: one b128 (float4) per iteration,
    // no VGPR round-trip, tracked by ASYNCcnt.
    const float* peb = pe + (size_t)b * EL * AT;
    for (int i = tid; i < EL * (AT / 4); i += EL) {
        const int s = i >> 5;                   // AT/4 = 32 quads per row
        const int q = i & 31;
        const float* gsrc = peb + (size_t)s * AT + (q << 2);
        const unsigned lds_off = (unsigned)((s * PE_STRIDE + (q << 2)) * sizeof(float));
        asm volatile("global_load_async_to_lds_b128 %0, %1, off"
                     :: "v"(lds_off), "v"(gsrc) : "memory");
    }
    for (int i = tid; i < TCHUNK * AT; i += EL)
        s_pd[i] = pd[(size_t)b * DL * AT + (size_t)(t0 + (i >> 7)) * AT + (i & (AT - 1))];
    if (tid < AT) s_v[tid] = vvec[tid];
    const float mval = mask[b * EL + tid];
    asm volatile("s_wait_asynccnt 0x0" ::: "memory");
    __syncthreads();

    const float* myrow = s_pe + tid * PE_STRIDE;
    for (int t = 0; t < TCHUNK; ++t) {
        const float* pdr = s_pd + t * AT;
        float acc = 0.f;
#pragma unroll 8
        for (int a = 0; a < AT; a += 4) {
            const float4 p = *(const float4*)(myrow + a);
            const float4 d = *(const float4*)(pdr + a);   // LDS broadcast
            const float4 vv = *(const float4*)(s_v + a);  // LDS broadcast
            acc += vv.x * TANHF(d.x + p.x);
            acc += vv.y * TANHF(d.y + p.y);
            acc += vv.z * TANHF(d.z + p.z);
            acc += vv.w * TANHF(d.w + p.w);
        }

        // Block max: wave32 shuffle reduce, then 16-partial LDS combine.
        float wm = wave_max(acc);
        if (lane == 0) s_red[wv] = wm;
        __syncthreads();
        float mx = s_red[0];
#pragma unroll
        for (int w = 1; w < NWAVES; ++w) mx = fmaxf(mx, s_red[w]);
        __syncthreads();

        const float p = __expf(acc - mx) * mval;

        // Block sum of masked exponentials.
        float ws = wave_sum(p);
        if (lane == 0) s_red[wv] = ws;
        __syncthreads();
        float denom = s_red[0];
#pragma unroll
        for (int w = 1; w < NWAVES; ++w) denom += s_red[w];
        __syncthreads();

        attn[(size_t)b * DL * EL + (size_t)(t0 + t) * EL + tid] = p / denom;
    }
}

// ---------------------------------------------------------------------------
// Stage 3: context GEMM   C[b] = attn[b] (256x512) @ E[b] (512x512)
// Same WMMA tiling; A loads are contiguous float2, B loads are two strided
// dwords (E is K-major here) with a prefetch hint on the next K panel.
// ---------------------------------------------------------------------------
__global__ void context_gemm_kernel(const float* __restrict__ Wt, // [B,DL,EL]
                                    const float* __restrict__ E,  // [B,EL,EH]
                                    float* __restrict__ C)        // [B,DL,EH]
{
    const int wave = threadIdx.x >> 5;
    const int lane = threadIdx.x & 31;
    const int tile = blockIdx.x * (blockDim.x >> 5) + wave;
    if (tile >= B_ * (DL / 16) * (EH / 16)) return;

    const int b  = tile >> 9;                 // 512 tiles per batch
    const int r  = tile & 511;
    const int mt = r >> 5;                    // EH/16 = 32 n-tiles
    const int nt = r & 31;
    const int row0 = mt << 4;
    const int col0 = nt << 4;
    const int m  = lane & 15;
    const int hi = lane >> 4;

    const float* Arow = Wt + (size_t)b * DL * EL + (size_t)(row0 + m) * EL;
    const float* Eb   = E + (size_t)b * EL * EH;

    v8f acc = {0.f, 0.f, 0.f, 0.f, 0.f, 0.f, 0.f, 0.f};
#pragma unroll 4
    for (int k0 = 0; k0 < EL; k0 += 4) {
        const int kk = k0 + (hi << 1);
        v2f a = *(const v2f*)(Arow + kk);
        v2f bb;
        bb.x = Eb[(size_t)kk * EH + col0 + m];
        bb.y = Eb[(size_t)(kk + 1) * EH + col0 + m];
        // Hint the next K panel of the strided B stream into cache.
        __builtin_prefetch(Eb + (size_t)(kk + 16) * EH + col0 + m, 0, 3);
        acc = __builtin_amdgcn_wmma_f32_16x16x4_f32(
            false, a, false, bb, (short)0, acc, false, false);
    }

    float* Crow = C + (size_t)b * DL * EH;
#pragma unroll
    for (int i = 0; i < 8; ++i) {
        const int rr = row0 + i + (hi << 3);
        Crow[(size_t)rr * EH + col0 + m] = acc[i];
    }
}

// ---------------------------------------------------------------------------
extern "C" void kernel_launch(void* const* d_in, const int* in_sizes, int n_in,
                              void* d_out, int out_size, void* d_ws, size_t ws_size,
                              hipStream_t stream)
{
    (void)in_sizes; (void)n_in; (void)out_size; (void)ws_size;

    const float* encoded = (const float*)d_in[0]; // [8,512,512]
    const float* decst   = (const float*)d_in[1]; // [8,256,512]
    const float* mask    = (const float*)d_in[2]; // [8,512]
    const float* Wh      = (const float*)d_in[3]; // [128,512]
    const float* Ws      = (const float*)d_in[4]; // [128,512]
    const float* Wsb     = (const float*)d_in[5]; // [128]
    const float* v       = (const float*)d_in[6]; // [128]

    float* context  = (float*)d_out;                         // [8,256,512]
    float* attn     = context + (size_t)B_ * DL * EH;        // [8,256,512]
    float* proj_enc = (float*)d_ws;                          // [8,512,128] = 2MB
    float* proj_dec = proj_enc + (size_t)B_ * EL * AT;       // [8,256,128] = 1MB

    // Stage 1: projections (WMMA f32 16x16x4).
    proj_gemm_kernel<<<2048 / 8, 256, 0, stream>>>(encoded, Wh, nullptr, proj_enc, 2048);
    proj_gemm_kernel<<<1024 / 8, 256, 0, stream>>>(decst, Ws, Wsb, proj_dec, 1024);

    // Stage 2: fused tanh scores + masked softmax (async-staged LDS pe).
    hipFuncSetAttribute((const void*)attn_score_softmax_kernel,
                        hipFuncAttributeMaxDynamicSharedMemorySize, SMEM_BYTES);
    attn_score_softmax_kernel<<<B_ * (DL / TCHUNK), EL, SMEM_BYTES, stream>>>(
        proj_enc, proj_dec, v, mask, attn);

    // Stage 3: context GEMM (WMMA f32 16x16x4). 4096 tiles / 8 waves = 512 blocks
    context_gemm_kernel<<<512, 256, 0, stream>>>(attn, encoded, context);
}